// PermutohedralX_5995774345316
// MI455X (gfx1250) — compile-verified
//
#include <hip/hip_runtime.h>
#include <hip/hip_bf16.h>
#include <stdint.h>

#ifndef __has_builtin
#define __has_builtin(x) 0
#endif

#define N_PTS   1048576
#define VD      21
#define DP1     6
#define M_LAT   2000000
#define ROWS    (M_LAT + 1)

static const size_t VALS_ELEMS        = (size_t)ROWS * VD;   // 42,000,021 floats (168 MB)
static const size_t VALS_STRIDE_ELEMS = 42000032ull;         // padded to 128B multiple

// ---------------------------------------------------------------- zero
__global__ __launch_bounds__(256) void k_zero(float* __restrict__ p, size_t n) {
    size_t i      = (size_t)blockIdx.x * blockDim.x + threadIdx.x;
    size_t stride = (size_t)gridDim.x * blockDim.x;
    for (; i < n; i += stride)
        __builtin_nontemporal_store(0.0f, &p[i]);            // no reuse before atomics
}

// ---------------------------------------------------------------- splat
// vals[os[n,j], v] += ws[n,j] * inp[n, v]   (one thread per (n,j,v) element)
__global__ __launch_bounds__(256) void k_splat(const float* __restrict__ inp,
                                               const float* __restrict__ ws,
                                               const int*   __restrict__ os,
                                               float*       __restrict__ vals) {
    unsigned gid = blockIdx.x * blockDim.x + threadIdx.x;   // total = N*6*21, exact grid
    unsigned n = gid / (DP1 * VD);
    unsigned r = gid - n * (DP1 * VD);
    unsigned j = r / VD;
    unsigned v = r - j * VD;
    int   o = os[(size_t)n * DP1 + j];                       // broadcast across 21 lanes
    float w = ws[(size_t)n * DP1 + j];
    // inp is read exactly once -> non-temporal, keep L2 free for the lattice
    float x = __builtin_nontemporal_load(&inp[(size_t)n * VD + v]);
    unsafeAtomicAdd(&vals[(size_t)o * VD + v], w * x);       // global_atomic_add_f32
}

// ---------------------------------------------------------------- blur (one pass)
// dst[0,:] = src[0,:] ; dst[i,:] = src[i,:] + 0.5*(src[n1[i-1],:] + src[n2[i-1],:])
// Grid sized so (gridDim*blockDim) % 21 == 0: lane keeps fixed column v, row strides.
// src (168 MB) is kept L2-resident (RT loads); dst is written non-temporally so the
// write stream cannot evict src lines (src alone fits the 192 MB L2, src+dst don't).
__global__ __launch_bounds__(256) void k_blur(const float* __restrict__ src,
                                              float*       __restrict__ dst,
                                              const int*   __restrict__ nbr /* [M][2] */) {
    unsigned tid = blockIdx.x * blockDim.x + threadIdx.x;   // 4095*256 = 21*49920 threads
    size_t   row = tid / VD;
    unsigned v   = tid - (unsigned)row * VD;
    const size_t rowStride = ((size_t)gridDim.x * blockDim.x) / VD;  // 49920

    for (; row < (size_t)ROWS; row += rowStride) {
        size_t idx = row * VD + v;

        // Software prefetch one iteration ahead: load next neighbor indices
        // (streamed, non-temporal) and prefetch their lattice rows into cache.
        size_t nrow = row + rowStride;
        if (v == 0 && nrow < (size_t)ROWS) {
            int pn1 = __builtin_nontemporal_load(&nbr[(nrow - 1) * 2 + 0]);
            int pn2 = __builtin_nontemporal_load(&nbr[(nrow - 1) * 2 + 1]);
            __builtin_prefetch(&src[(size_t)pn1 * VD], 0, 3);   // global_prefetch_b8
            __builtin_prefetch(&src[(size_t)pn2 * VD], 0, 3);
        }

        float s = src[idx];                                  // RT: vals stays L2-resident
        if (row == 0) { __builtin_nontemporal_store(s, &dst[idx]); continue; }
        int n1 = __builtin_nontemporal_load(&nbr[(row - 1) * 2 + 0]);
        int n2 = __builtin_nontemporal_load(&nbr[(row - 1) * 2 + 1]);
        float a = src[(size_t)n1 * VD + v];
        float b = src[(size_t)n2 * VD + v];
        __builtin_nontemporal_store(s + 0.5f * (a + b), &dst[idx]);
    }
}

// ---------------------------------------------------------------- slice
// out[n,v] = alpha * sum_j ws[n,j] * vals[os[n,j], v]
// Per-block os/ws tiles staged into LDS via gfx1250 async global->LDS loads.
#define PTS_PER_BLK 64

#if __has_builtin(__builtin_amdgcn_global_load_async_to_lds_b32)
#define HAVE_ASYNC_LDS 1
#else
#define HAVE_ASYNC_LDS 0
#endif

__global__ __launch_bounds__(256) void k_slice(const float* __restrict__ ws,
                                               const int*   __restrict__ os,
                                               const float* __restrict__ vals,
                                               float*       __restrict__ out) {
    __shared__ int   os_s[PTS_PER_BLK * DP1];   // 384 ints
    __shared__ float ws_s[PTS_PER_BLK * DP1];   // 384 floats
    const unsigned tid  = threadIdx.x;
    const unsigned base = blockIdx.x * PTS_PER_BLK;          // first point of block
    const unsigned nIdx = PTS_PER_BLK * DP1;                 // 384

#if HAVE_ASYNC_LDS
    for (unsigned i = tid; i < nIdx; i += blockDim.x) {
        // (global src, LDS dst, imm offset, imm cpol) -- matches global_load_lds convention
        __builtin_amdgcn_global_load_async_to_lds_b32(
            (int*)(os + (size_t)base * DP1 + i),
            (int*)&os_s[i], 0, 0);
        __builtin_amdgcn_global_load_async_to_lds_b32(
            (int*)(ws + (size_t)base * DP1 + i),
            (int*)&ws_s[i], 0, 0);
    }
#if __has_builtin(__builtin_amdgcn_s_wait_asynccnt)
    __builtin_amdgcn_s_wait_asynccnt(0);
#else
    asm volatile("s_wait_asynccnt 0" ::: "memory");
#endif
    __syncthreads();
#else
    for (unsigned i = tid; i < nIdx; i += blockDim.x) {
        os_s[i] = __builtin_nontemporal_load(&os[(size_t)base * DP1 + i]);
        ws_s[i] = __builtin_nontemporal_load(&ws[(size_t)base * DP1 + i]);
    }
    __syncthreads();
#endif

    const float alpha = 32.0f / 33.0f;                       // 1/(1+2^-5), d = 5
    const unsigned totalE = PTS_PER_BLK * VD;                // 1344 elements per block
    for (unsigned e = tid; e < totalE; e += blockDim.x) {
        unsigned p = e / VD;
        unsigned v = e - p * VD;
        float acc = 0.0f;
#pragma unroll
        for (int j = 0; j < DP1; ++j) {
            int   o = os_s[p * DP1 + j];
            float w = ws_s[p * DP1 + j];
            acc += w * vals[(size_t)o * VD + v];             // RT gather, mostly L2 hits
        }
        __builtin_nontemporal_store(alpha * acc, &out[(size_t)base * VD + e]);
    }
}

// ---------------------------------------------------------------- launch
extern "C" void kernel_launch(void* const* d_in, const int* in_sizes, int n_in,
                              void* d_out, int out_size, void* d_ws, size_t ws_size,
                              hipStream_t stream) {
    const float* inp = (const float*)d_in[0];   // (N, 21)
    const float* ws  = (const float*)d_in[1];   // (N, 6)
    const int*   os  = (const int*)  d_in[2];   // (N, 6)
    const int*   nbr = (const int*)  d_in[3];   // (6, M, 2)
    float*       out = (float*)d_out;           // (N, 21)

    float* vals0 = (float*)d_ws;
    float* vals1 = vals0 + VALS_STRIDE_ELEMS;   // 2 x 168 MB ping-pong in scratch

    // 1) zero the splat target
    k_zero<<<8192, 256, 0, stream>>>(vals0, VALS_ELEMS);

    // 2) splat: one thread per (n, j, v); exact grid (132,120,576 / 256)
    k_splat<<<(N_PTS * DP1 * VD) / 256, 256, 0, stream>>>(inp, ws, os, vals0);

    // 3) 6 blur passes, double-buffered (reference reads old vals on the RHS)
    float* src = vals0;
    float* dst = vals1;
    for (int j = 0; j < DP1; ++j) {
        k_blur<<<4095, 256, 0, stream>>>(src, dst, nbr + (size_t)j * M_LAT * 2);
        float* t = src; src = dst; dst = t;
    }
    // after an even number of passes, result is back in vals0 (== src)

    // 4) slice
    k_slice<<<N_PTS / PTS_PER_BLK, 256, 0, stream>>>(ws, os, src, out);
}